// Attention_MSF_5592047420192
// MI455X (gfx1250) — compile-verified
//
#include <hip/hip_runtime.h>
#include <math.h>

// ---------------------------------------------------------------------------
// Problem constants (from reference): B=4, N=2048, DIM=256, G=2, G_DIM=128,
// G_H=4, HD=32, SCALE=32^-0.5, NUM_NEI=(16,32), CHANNEL=128, D_MSF=128.
// ---------------------------------------------------------------------------
#define BATCH   4
#define NPTS    2048
#define DIM     256
#define QKVD    768
#define GDIM    128
#define SCALE_F 0.17677669529663689f   // 1/sqrt(32)

typedef float v2f __attribute__((ext_vector_type(2)));
typedef float v8f __attribute__((ext_vector_type(8)));

__device__ __forceinline__ float geluf(float v) {
    // exact gelu: 0.5*x*(1+erf(x/sqrt(2)))
    return 0.5f * v * (1.0f + erff(v * 0.7071067811865476f));
}

// ---------------------------------------------------------------------------
// Kernel 1: KNN — one 256-thread block per query point.
// Computes 2048 squared distances in LDS, then 32 rounds of block argmin on
// packed (dist_bits, idx) keys. First 16 selected = 16-NN, all 32 = 32-NN.
// ---------------------------------------------------------------------------
__global__ __launch_bounds__(256) void knn_kernel(const float* __restrict__ pos,
                                                  int* __restrict__ knn) {
    __shared__ float px[NPTS], py[NPTS], pz[NPTS], dist[NPTS];
    __shared__ unsigned long long red[8];
    const int tid = threadIdx.x;
    const int bn  = blockIdx.x;          // 0..8191
    const int b   = bn >> 11;
    const int n   = bn & (NPTS - 1);
    const float* pb = pos + (size_t)b * NPTS * 3;
    for (int i = tid; i < NPTS; i += 256) {
        px[i] = pb[i * 3 + 0];
        py[i] = pb[i * 3 + 1];
        pz[i] = pb[i * 3 + 2];
    }
    __syncthreads();
    const float qx = px[n], qy = py[n], qz = pz[n];
    for (int i = tid; i < NPTS; i += 256) {
        float dx = px[i] - qx, dy = py[i] - qy, dz = pz[i] - qz;
        dist[i] = dx * dx + dy * dy + dz * dz;
    }
    __syncthreads();
    int* outp = knn + (size_t)bn * 32;
    for (int it = 0; it < 32; ++it) {
        unsigned long long best = ~0ull;
        #pragma unroll
        for (int c = 0; c < 8; ++c) {
            int i = tid * 8 + c;
            unsigned long long key =
                ((unsigned long long)__float_as_uint(dist[i]) << 32) | (unsigned)i;
            best = (key < best) ? key : best;
        }
        #pragma unroll
        for (int off = 16; off > 0; off >>= 1) {
            unsigned hi = __shfl_xor((unsigned)(best >> 32), off, 32);
            unsigned lo = __shfl_xor((unsigned)(best & 0xffffffffu), off, 32);
            unsigned long long o = (((unsigned long long)hi) << 32) | lo;
            best = (o < best) ? o : best;
        }
        if ((tid & 31) == 0) red[tid >> 5] = best;
        __syncthreads();
        if (tid == 0) {
            unsigned long long w = red[0];
            #pragma unroll
            for (int q = 1; q < 8; ++q) w = (red[q] < w) ? red[q] : w;
            int widx = (int)(w & 0xffffffffu);
            outp[it] = widx;
            dist[widx] = __uint_as_float(0x7f800000u);   // +inf: mask out
        }
        __syncthreads();
    }
}

// ---------------------------------------------------------------------------
// Kernel 2/4/7: fp32 WMMA GEMM, C[8192,NOUT] = A[8192,256] @ B[256,NOUT].
// One wave computes a 32x32 tile via V_WMMA_F32_16X16X4_F32 (4 accs, 64 K-steps).
// MODE 0: qkv (no bias).  MODE 1: gelu(acc+bias) -> feats_proj.
// MODE 2: A scaled per (batch,k) by avs; out = extra + bias + acc (head GEMM).
// ---------------------------------------------------------------------------
template <int NOUT, int MODE>
__global__ __launch_bounds__(256) void gemm_kernel(const float* __restrict__ A,
                                                   const float* __restrict__ B,
                                                   const float* __restrict__ bias,
                                                   const float* __restrict__ extra,
                                                   const float* __restrict__ avs,
                                                   float* __restrict__ C) {
    const int tilesM = 8192 / 32;
    const int waveId = (int)((blockIdx.x * blockDim.x + threadIdx.x) >> 5);
    const int lane   = threadIdx.x & 31;
    const int tm = waveId % tilesM;
    const int tn = waveId / tilesM;
    const int m0 = tm * 32, n0 = tn * 32;
    const int r    = lane & 15;
    const int kq2  = (lane >> 4) << 1;          // 0 or 2
    const float* Ar0 = A + (size_t)(m0 + r) * DIM;
    const float* Ar1 = A + (size_t)(m0 + 16 + r) * DIM;
    const float* avrow = (MODE == 2) ? (avs + (size_t)(m0 >> 11) * DIM) : nullptr;
    v8f acc00 = {}, acc01 = {}, acc10 = {}, acc11 = {};
    const int c0 = n0 + r, c1 = n0 + 16 + r;
    #pragma unroll 4
    for (int k0 = 0; k0 < DIM; k0 += 4) {
        const int kc = k0 + kq2;
        float2 t0 = *(const float2*)(Ar0 + kc);
        float2 t1 = *(const float2*)(Ar1 + kc);
        if (MODE == 2) {
            float2 sc = *(const float2*)(avrow + kc);
            t0.x *= sc.x; t0.y *= sc.y;
            t1.x *= sc.x; t1.y *= sc.y;
        }
        v2f a0; a0[0] = t0.x; a0[1] = t0.y;
        v2f a1; a1[0] = t1.x; a1[1] = t1.y;
        const float* Bk = B + (size_t)kc * NOUT;
        v2f b0; b0[0] = Bk[c0]; b0[1] = Bk[NOUT + c0];
        v2f b1; b1[0] = Bk[c1]; b1[1] = Bk[NOUT + c1];
        acc00 = __builtin_amdgcn_wmma_f32_16x16x4_f32(false, a0, false, b0, (short)0, acc00, false, false);
        acc01 = __builtin_amdgcn_wmma_f32_16x16x4_f32(false, a0, false, b1, (short)0, acc01, false, false);
        acc10 = __builtin_amdgcn_wmma_f32_16x16x4_f32(false, a1, false, b0, (short)0, acc10, false, false);
        acc11 = __builtin_amdgcn_wmma_f32_16x16x4_f32(false, a1, false, b1, (short)0, acc11, false, false);
    }
    const int rowOff = (lane >> 4) * 8;
    const float bc0 = (MODE == 0) ? 0.f : bias[c0];
    const float bc1 = (MODE == 0) ? 0.f : bias[c1];
    #pragma unroll
    for (int e = 0; e < 8; ++e) {
        const int row0 = m0 + rowOff + e;
        const int row1 = m0 + 16 + rowOff + e;
        float o00 = acc00[e] + bc0, o01 = acc01[e] + bc1;
        float o10 = acc10[e] + bc0, o11 = acc11[e] + bc1;
        if (MODE == 1) { o00 = geluf(o00); o01 = geluf(o01); o10 = geluf(o10); o11 = geluf(o11); }
        if (MODE == 2) {
            o00 += extra[(size_t)row0 * NOUT + c0]; o01 += extra[(size_t)row0 * NOUT + c1];
            o10 += extra[(size_t)row1 * NOUT + c0]; o11 += extra[(size_t)row1 * NOUT + c1];
        }
        C[(size_t)row0 * NOUT + c0] = o00;
        C[(size_t)row0 * NOUT + c1] = o01;
        C[(size_t)row1 * NOUT + c0] = o10;
        C[(size_t)row1 * NOUT + c1] = o11;
    }
}

// ---------------------------------------------------------------------------
// Kernel 3: neighborhood attention for one head-group. One wave per (b,n);
// lane owns 4 of the 128 group dims (head = lane/8, width-8 shfl reductions).
// Pass 1 computes logits (q.k*scale + sum(gelu rel-MLP)); pass 2 recomputes
// the rel-MLP and accumulates softmax-weighted (v + vrp).
// ---------------------------------------------------------------------------
template <int NK>
__global__ __launch_bounds__(256) void attn_kernel(const float* __restrict__ qkv,
                                                   const float* __restrict__ pos,
                                                   const int* __restrict__ knn,
                                                   const float* __restrict__ Wp,
                                                   const float* __restrict__ bp,
                                                   float* __restrict__ xcat,
                                                   int gOff) {
    const int wave = (int)((blockIdx.x * blockDim.x + threadIdx.x) >> 5);
    const int lane = threadIdx.x & 31;
    const int bn = wave;                 // 0..8191
    const int b  = bn >> 11;
    const int d0 = lane * 4;
    const float4 q4  = *(const float4*)(qkv + (size_t)bn * QKVD + gOff + d0);
    const float4 wp0 = *(const float4*)(Wp + d0);
    const float4 wp1 = *(const float4*)(Wp + GDIM + d0);
    const float4 wp2 = *(const float4*)(Wp + 2 * GDIM + d0);
    const float4 bp4 = *(const float4*)(bp + d0);
    const float* pq = pos + (size_t)bn * 3;
    const float qx = pq[0], qy = pq[1], qz = pq[2];
    const int* kb = knn + (size_t)bn * 32;
    const size_t bBase = (size_t)b * NPTS;

    float lg[NK];
    int   idxs[NK];
    #pragma unroll
    for (int j = 0; j < NK; ++j) {
        const int idx = kb[j];
        idxs[j] = idx;
        const float* pj = pos + (bBase + idx) * 3;
        const float rx = qx - pj[0], ry = qy - pj[1], rz = qz - pj[2];
        const float4 k4 = *(const float4*)(qkv + (bBase + idx) * QKVD + 256 + gOff + d0);
        const float w0 = geluf(rx * wp0.x + ry * wp1.x + rz * wp2.x + bp4.x);
        const float w1 = geluf(rx * wp0.y + ry * wp1.y + rz * wp2.y + bp4.y);
        const float w2 = geluf(rx * wp0.z + ry * wp1.z + rz * wp2.z + bp4.z);
        const float w3 = geluf(rx * wp0.w + ry * wp1.w + rz * wp2.w + bp4.w);
        float p = (q4.x * k4.x + q4.y * k4.y + q4.z * k4.z + q4.w * k4.w) * SCALE_F
                  + (w0 + w1 + w2 + w3);
        p += __shfl_xor(p, 1, 8);
        p += __shfl_xor(p, 2, 8);
        p += __shfl_xor(p, 4, 8);
        lg[j] = p;
    }
    float mx = lg[0];
    #pragma unroll
    for (int j = 1; j < NK; ++j) mx = fmaxf(mx, lg[j]);
    float s = 0.f;
    #pragma unroll
    for (int j = 0; j < NK; ++j) { float e = expf(lg[j] - mx); lg[j] = e; s += e; }
    const float inv = 1.f / s;

    float4 acc = make_float4(0.f, 0.f, 0.f, 0.f);
    #pragma unroll
    for (int j = 0; j < NK; ++j) {
        const int idx = idxs[j];
        const float* pj = pos + (bBase + idx) * 3;
        const float rx = qx - pj[0], ry = qy - pj[1], rz = qz - pj[2];
        const float4 v4 = *(const float4*)(qkv + (bBase + idx) * QKVD + 512 + gOff + d0);
        const float a = lg[j] * inv;
        acc.x += a * (v4.x + geluf(rx * wp0.x + ry * wp1.x + rz * wp2.x + bp4.x));
        acc.y += a * (v4.y + geluf(rx * wp0.y + ry * wp1.y + rz * wp2.y + bp4.y));
        acc.z += a * (v4.z + geluf(rx * wp0.z + ry * wp1.z + rz * wp2.z + bp4.z));
        acc.w += a * (v4.w + geluf(rx * wp0.w + ry * wp1.w + rz * wp2.w + bp4.w));
    }
    *(float4*)(xcat + (size_t)bn * DIM + gOff + d0) = acc;
}

// ---------------------------------------------------------------------------
// Kernel 5: feats_S[b][d] = mean over N of feats_proj. One block per batch,
// thread t owns channel t (coalesced column reduction).
// ---------------------------------------------------------------------------
__global__ __launch_bounds__(256) void mean_kernel(const float* __restrict__ fp,
                                                   float* __restrict__ fS) {
    const int b = blockIdx.x, t = threadIdx.x;
    const float* base = fp + (size_t)b * NPTS * DIM + t;
    float s = 0.f;
    for (int n = 0; n < NPTS; ++n) s += base[(size_t)n * DIM];
    fS[b * DIM + t] = s * (1.0f / (float)NPTS);
}

// ---------------------------------------------------------------------------
// Kernel 6: tiny MLP: z=gelu(fS@Wfc1+b), av=z@Wfc2+b, softmax over M_MSF=2.
// Single block (256 threads), stages separated by LDS barriers.
// ---------------------------------------------------------------------------
__global__ __launch_bounds__(256) void mlp_kernel(const float* __restrict__ fS,
                                                  const float* __restrict__ Wfc1,
                                                  const float* __restrict__ bfc1,
                                                  const float* __restrict__ Wfc2,
                                                  const float* __restrict__ bfc2,
                                                  float* __restrict__ avs) {
    __shared__ float z[BATCH * 128];
    __shared__ float av[BATCH * 256];
    const int t = threadIdx.x;
    for (int o = t; o < BATCH * 128; o += 256) {
        const int b = o >> 7, h = o & 127;
        float s = bfc1[h];
        for (int k = 0; k < DIM; ++k) s += fS[b * DIM + k] * Wfc1[k * 128 + h];
        z[o] = geluf(s);
    }
    __syncthreads();
    for (int o = t; o < BATCH * 256; o += 256) {
        const int b = o >> 8, d = o & 255;
        float s = bfc2[d];
        for (int k = 0; k < 128; ++k) s += z[b * 128 + k] * Wfc2[k * 256 + d];
        av[o] = s;
    }
    __syncthreads();
    for (int o = t; o < BATCH * 128; o += 256) {
        const int b = o >> 7, c = o & 127;
        const float a0 = av[b * 256 + c];
        const float a1 = av[b * 256 + 128 + c];
        const float m = fmaxf(a0, a1);
        const float e0 = expf(a0 - m), e1 = expf(a1 - m);
        const float inv = 1.f / (e0 + e1);
        avs[b * 256 + c]       = e0 * inv;
        avs[b * 256 + 128 + c] = e1 * inv;
    }
}

// ---------------------------------------------------------------------------
// Orchestration. ws layout (bytes):
//   [0, 1MB)          knn  int[8192*32]
//   [1MB, +24MB)      qkv  float[8192*768]
//   [+,  +8MB)        xcat float[8192*256]
//   [+,  +8MB)        fproj float[8192*256]
//   [+,  4KB)         fS   float[4*256]
//   [+,  4KB)         avs  float[4*256]
// ---------------------------------------------------------------------------
extern "C" void kernel_launch(void* const* d_in, const int* in_sizes, int n_in,
                              void* d_out, int out_size, void* d_ws, size_t ws_size,
                              hipStream_t stream) {
    const float* x      = (const float*)d_in[0];
    const float* pos    = (const float*)d_in[1];
    const float* Wqkv   = (const float*)d_in[2];
    const float* Wp0    = (const float*)d_in[3];
    const float* bp0    = (const float*)d_in[4];
    const float* Wp1    = (const float*)d_in[5];
    const float* bp1    = (const float*)d_in[6];
    const float* W_proj = (const float*)d_in[7];
    const float* b_proj = (const float*)d_in[8];
    const float* W_fc1  = (const float*)d_in[9];
    const float* b_fc1  = (const float*)d_in[10];
    const float* W_fc2  = (const float*)d_in[11];
    const float* b_fc2  = (const float*)d_in[12];
    const float* W_head = (const float*)d_in[13];
    const float* b_head = (const float*)d_in[14];
    float* out = (float*)d_out;

    char* ws = (char*)d_ws;
    size_t o = 0;
    int*   knn   = (int*)(ws + o);   o += (size_t)8192 * 32 * sizeof(int);      // 1 MB
    float* qkv   = (float*)(ws + o); o += (size_t)8192 * QKVD * sizeof(float);  // 24 MB
    float* xcat  = (float*)(ws + o); o += (size_t)8192 * DIM * sizeof(float);   // 8 MB
    float* fproj = (float*)(ws + o); o += (size_t)8192 * DIM * sizeof(float);   // 8 MB
    float* fS    = (float*)(ws + o); o += (size_t)BATCH * DIM * sizeof(float);
    float* avs   = (float*)(ws + o); o += (size_t)BATCH * DIM * sizeof(float);

    // 1. KNN (top-32 per query; first 16 are the 16-NN)
    knn_kernel<<<BATCH * NPTS, 256, 0, stream>>>(pos, knn);

    // 2. qkv = x @ Wqkv   (WMMA fp32; 8192x256x768)
    gemm_kernel<QKVD, 0><<<(8192 / 32) * (QKVD / 32) / 8, 256, 0, stream>>>(
        x, Wqkv, nullptr, nullptr, nullptr, qkv);

    // 3. attention, both groups -> xcat
    attn_kernel<16><<<BATCH * NPTS / 8, 256, 0, stream>>>(qkv, pos, knn, Wp0, bp0, xcat, 0);
    attn_kernel<32><<<BATCH * NPTS / 8, 256, 0, stream>>>(qkv, pos, knn, Wp1, bp1, xcat, GDIM);

    // 4. feats_proj = gelu(xcat @ W_proj + b_proj)
    gemm_kernel<DIM, 1><<<(8192 / 32) * (DIM / 32) / 8, 256, 0, stream>>>(
        xcat, W_proj, b_proj, nullptr, nullptr, fproj);

    // 5. feats_S = mean_N(feats_proj)
    mean_kernel<<<BATCH, 256, 0, stream>>>(fproj, fS);

    // 6. tiny MLP + group softmax -> avs
    mlp_kernel<<<1, 256, 0, stream>>>(fS, W_fc1, b_fc1, W_fc2, b_fc2, avs);

    // 7. out = fproj + (xcat * avs) @ W_head + b_head
    gemm_kernel<DIM, 2><<<(8192 / 32) * (DIM / 32) / 8, 256, 0, stream>>>(
        xcat, W_head, b_head, fproj, avs, out);
}